// SparseSegmentationNet_66340064854706
// MI455X (gfx1250) — compile-verified
//
#include <hip/hip_runtime.h>
#include <hip/hip_bf16.h>

typedef __attribute__((ext_vector_type(2))) float v2f;
typedef __attribute__((ext_vector_type(8))) float v8f;

#define DGRID 128
#define DCUBE (DGRID * DGRID * DGRID)

// ---------------------------------------------------------------------------
// Phase 1: dense hash grid (lin coord -> point index, -1 if empty)
// ---------------------------------------------------------------------------
__global__ void init_grid_kernel(int* __restrict__ grid, int n) {
    int i = blockIdx.x * blockDim.x + threadIdx.x;
    if (i < n) grid[i] = -1;
}

__global__ void scatter_kernel(const int* __restrict__ coords, int* __restrict__ grid, int n) {
    int i = blockIdx.x * blockDim.x + threadIdx.x;
    if (i >= n) return;
    int x = coords[3 * i + 0];
    int y = coords[3 * i + 1];
    int z = coords[3 * i + 2];
    grid[x * DGRID * DGRID + y * DGRID + z] = i;
}

// ---------------------------------------------------------------------------
// Phase 2: rulebook, k-major layout nbr[k*N + p] for coalesced access
// k = (dx+1)*9 + (dy+1)*3 + (dz+1)  (matches K_OFFSETS ordering)
// ---------------------------------------------------------------------------
__global__ void rulebook_kernel(const int* __restrict__ coords, const int* __restrict__ grid,
                                int* __restrict__ nbr, int n) {
    int i = blockIdx.x * blockDim.x + threadIdx.x;
    if (i >= n) return;
    int x = coords[3 * i + 0];
    int y = coords[3 * i + 1];
    int z = coords[3 * i + 2];
#pragma unroll
    for (int dx = -1; dx <= 1; ++dx)
#pragma unroll
        for (int dy = -1; dy <= 1; ++dy)
#pragma unroll
            for (int dz = -1; dz <= 1; ++dz) {
                int k = (dx + 1) * 9 + (dy + 1) * 3 + (dz + 1);
                int nx = x + dx, ny = y + dy, nz = z + dz;
                int v = -1;
                if (nx >= 0 && nx < DGRID && ny >= 0 && ny < DGRID && nz >= 0 && nz < DGRID)
                    v = grid[nx * DGRID * DGRID + ny * DGRID + nz];
                nbr[k * n + i] = v;
            }
}

// ---------------------------------------------------------------------------
// Weight pre-pack: reorder w[k][cin][cout] into the exact per-lane WMMA
// B-operand layout so the conv kernels fetch one contiguous float2 per lane:
//   Wp[((k*KC + kc)*NT + nt)*32 + lane][v]  =  w[k][kc*4 + 2*g + v][nt*16 + m]
// with lane = g*16 + m.  One thread per packed element; runs once per call.
// ---------------------------------------------------------------------------
__global__ void pack_weights_kernel(const float* __restrict__ W, float* __restrict__ Wp,
                                    int CIN, int COUT) {
    int total = 27 * CIN * COUT;
    int i = blockIdx.x * blockDim.x + threadIdx.x;
    if (i >= total) return;
    int KC = CIN / 4;
    int NT = COUT / 16;
    int v = i & 1;
    int lane = (i >> 1) & 31;
    int rest = i >> 6;
    int nt = rest % NT;
    rest /= NT;
    int kc = rest % KC;
    int k = rest / KC;
    int g = lane >> 4;
    int m = lane & 15;
    int kk = kc * 4 + 2 * g + v;
    int nn = nt * 16 + m;
    Wp[i] = W[(k * CIN + kk) * COUT + nn];
}

// ---------------------------------------------------------------------------
// Phase 3: conv1 (1 -> 16). Tiny; pure VALU, one thread per point.
// ---------------------------------------------------------------------------
__global__ void conv1_kernel(const float* __restrict__ feats, const float* __restrict__ w1,
                             const int* __restrict__ nbr, float* __restrict__ x1, int n) {
    int i = blockIdx.x * blockDim.x + threadIdx.x;
    if (i >= n) return;
    float acc[16];
#pragma unroll
    for (int c = 0; c < 16; ++c) acc[c] = 0.f;
    for (int k = 0; k < 27; ++k) {
        int idx = nbr[k * n + i];
        int safe = idx < 0 ? 0 : idx;
        float g = feats[safe];
        if (idx < 0) g = 0.f;
#pragma unroll
        for (int c = 0; c < 16; ++c) acc[c] += g * w1[k * 16 + c];
    }
#pragma unroll
    for (int c = 0; c < 16; ++c) x1[(long)i * 16 + c] = acc[c];
}

// ---------------------------------------------------------------------------
// Phase 4/5: gather-GEMM sparse conv via V_WMMA_F32_16X16X4_F32 (full fp32).
// One wave32 owns TWO 16-point M-subtiles (32 points) so each B register pair
// (loaded as a single coalesced b64 from the pre-packed layout) feeds two
// WMMAs.  Per k offset:
//   A (16x4):  gathered input rows. Lanes 0-15 hold K = 2g+{0,1} in the two
//              A-VGPRs, lanes 16-31 hold K = 2g+{0,1} = {2,3}; row M = lane%16.
//   B (4x16):  pre-packed W[k] slice, one float2 per lane.
//   C (16x16): f32, 8 VGPRs; NT = COUT/16 tiles per subtile in registers.
// EXEC stays all-ones through every WMMA: invalid neighbors use clamp+select;
// tail tiles only diverge at the final (non-WMMA) store.
// ---------------------------------------------------------------------------
template <int CIN, int COUT>
__global__ __launch_bounds__(256) void conv_wmma_kernel(const float* __restrict__ xin,
                                                        const float* __restrict__ Wp,
                                                        const int* __restrict__ nbr,
                                                        float* __restrict__ xout,
                                                        int n, int nTiles32) {
    constexpr int KC = CIN / 4;    // K chunks of 4
    constexpr int NT = COUT / 16;  // N tiles of 16
    int lane = threadIdx.x & 31;
    int wave = threadIdx.x >> 5;
    int tile = blockIdx.x * (blockDim.x >> 5) + wave;
    if (tile >= nTiles32) return;  // wave-uniform exit, EXEC stays full below

    int m = lane & 15;  // A row / B,C column
    int g = lane >> 4;  // half-wave K group
    int base = tile * 32;
    int pt0 = base + m;
    int pt1 = base + 16 + m;

    const v2f* bp = (const v2f*)Wp;

    v8f acc0[NT] = {};
    v8f acc1[NT] = {};

    for (int k = 0; k < 27; ++k) {
        int idx0 = (pt0 < n) ? nbr[k * n + pt0] : -1;
        int idx1 = (pt1 < n) ? nbr[k * n + pt1] : -1;
        const float* arow0 = xin + (long)(idx0 < 0 ? 0 : idx0) * CIN;
        const float* arow1 = xin + (long)(idx1 < 0 ? 0 : idx1) * CIN;
#pragma unroll
        for (int kc = 0; kc < KC; ++kc) {
            int kk = kc * 4 + 2 * g;
            v2f a0 = *(const v2f*)(arow0 + kk);  // 8B-aligned contiguous pair
            v2f a1 = *(const v2f*)(arow1 + kk);
            if (idx0 < 0) { a0.x = 0.f; a0.y = 0.f; }
            if (idx1 < 0) { a1.x = 0.f; a1.y = 0.f; }
#pragma unroll
            for (int nt = 0; nt < NT; ++nt) {
                v2f bv = bp[(size_t)(((k * KC + kc) * NT + nt)) * 32 + lane];
                acc0[nt] = __builtin_amdgcn_wmma_f32_16x16x4_f32(
                    false, a0, false, bv, (short)0, acc0[nt], false, false);
                acc1[nt] = __builtin_amdgcn_wmma_f32_16x16x4_f32(
                    false, a1, false, bv, (short)0, acc1[nt], false, false);
            }
        }
    }

    // C/D layout: VGPR r, lanes 0-15 -> row M=r, lanes 16-31 -> row M=8+r, col N=lane%16
#pragma unroll
    for (int nt = 0; nt < NT; ++nt) {
#pragma unroll
        for (int r = 0; r < 8; ++r) {
            int row0 = base + 8 * g + r;
            int row1 = base + 16 + 8 * g + r;
            if (row0 < n) xout[(long)row0 * COUT + nt * 16 + m] = acc0[nt][r];
            if (row1 < n) xout[(long)row1 * COUT + nt * 16 + m] = acc1[nt][r];
        }
    }
}

// ---------------------------------------------------------------------------
// Phase 6: head, out = x3 @ wm(64x3) + bm. One thread per point.
// ---------------------------------------------------------------------------
__global__ void head_kernel(const float* __restrict__ x3, const float* __restrict__ wm,
                            const float* __restrict__ bm, float* __restrict__ out, int n) {
    int i = blockIdx.x * blockDim.x + threadIdx.x;
    if (i >= n) return;
    float a0 = bm[0], a1 = bm[1], a2 = bm[2];
    const float* row = x3 + (long)i * 64;
#pragma unroll
    for (int c = 0; c < 64; ++c) {
        float v = row[c];
        a0 += v * wm[c * 3 + 0];
        a1 += v * wm[c * 3 + 1];
        a2 += v * wm[c * 3 + 2];
    }
    out[3 * i + 0] = a0;
    out[3 * i + 1] = a1;
    out[3 * i + 2] = a2;
}

// ---------------------------------------------------------------------------
extern "C" void kernel_launch(void* const* d_in, const int* in_sizes, int n_in,
                              void* d_out, int out_size, void* d_ws, size_t ws_size,
                              hipStream_t stream) {
    const int* coords = (const int*)d_in[0];
    const float* feats = (const float*)d_in[1];
    const float* w1 = (const float*)d_in[2];
    const float* w2 = (const float*)d_in[3];
    const float* w3 = (const float*)d_in[4];
    const float* wm = (const float*)d_in[5];
    const float* bm = (const float*)d_in[6];
    float* out = (float*)d_out;

    int n = in_sizes[0] / 3;  // N_PTS

    // scratch layout (~120 MB total)
    char* ws = (char*)d_ws;
    auto align256 = [](size_t x) { return (x + 255) & ~(size_t)255; };
    size_t off = 0;
    int* grid = (int*)(ws + off);
    off = align256(off + (size_t)DCUBE * 4);
    int* nbr = (int*)(ws + off);
    off = align256(off + (size_t)27 * n * 4);
    float* x1 = (float*)(ws + off);
    off = align256(off + (size_t)n * 16 * 4);
    float* x2 = (float*)(ws + off);
    off = align256(off + (size_t)n * 32 * 4);
    float* x3 = (float*)(ws + off);
    off = align256(off + (size_t)n * 64 * 4);
    float* wp2 = (float*)(ws + off);
    off = align256(off + (size_t)27 * 16 * 32 * 4);
    float* wp3 = (float*)(ws + off);
    off = align256(off + (size_t)27 * 32 * 64 * 4);
    (void)ws_size;
    (void)n_in;
    (void)out_size;

    const int tb = 256;
    init_grid_kernel<<<(DCUBE + tb - 1) / tb, tb, 0, stream>>>(grid, DCUBE);
    scatter_kernel<<<(n + tb - 1) / tb, tb, 0, stream>>>(coords, grid, n);
    rulebook_kernel<<<(n + tb - 1) / tb, tb, 0, stream>>>(coords, grid, nbr, n);

    pack_weights_kernel<<<(27 * 16 * 32 + tb - 1) / tb, tb, 0, stream>>>(w2, wp2, 16, 32);
    pack_weights_kernel<<<(27 * 32 * 64 + tb - 1) / tb, tb, 0, stream>>>(w3, wp3, 32, 64);

    conv1_kernel<<<(n + tb - 1) / tb, tb, 0, stream>>>(feats, w1, nbr, x1, n);

    int nTiles32 = (n + 31) / 32;
    int wavesPerBlock = tb / 32;
    int blocks = (nTiles32 + wavesPerBlock - 1) / wavesPerBlock;
    conv_wmma_kernel<16, 32><<<blocks, tb, 0, stream>>>(x1, wp2, nbr, x2, n, nTiles32);
    conv_wmma_kernel<32, 64><<<blocks, tb, 0, stream>>>(x2, wp3, nbr, x3, n, nTiles32);

    head_kernel<<<(n + tb - 1) / tb, tb, 0, stream>>>(x3, wm, bm, out, n);
}